// PlanNet_17772574670990
// MI455X (gfx1250) — compile-verified
//
#include <hip/hip_runtime.h>

// ---------------------------------------------------------------------------
// PlanNet GNN forward on MI455X (gfx1250, wave32).
// All GEMMs mapped to V_WMMA_F32_16X16X4_F32 (fp32 — matches reference
// numerics; compute-bound: all state fits in the 192MB L2).
// B operands are kept in a pair-swizzled layout  Wsw[k/2][col] = {W[k][col],
// W[k+1][col]}  so each lane's B fragment is a single aligned 8-byte load
// into an even register pair (no v_mov packing between WMMAs).
// ---------------------------------------------------------------------------

typedef __attribute__((ext_vector_type(2))) float v2f;
typedef __attribute__((ext_vector_type(8))) float v8f;

#define N_PATHS 65536
#define N_LINKS 16384
#define N_NODES 4096
#define PLEN    8
#define PD      32
#define LD      32
#define ND      32
#define RU      256
#define GN      96   /* 3*PD */

__device__ __forceinline__ v8f wmma4(v2f a, v2f b, v8f c) {
  // D(16x16,f32) += A(16x4,f32) * B(4x16,f32)
  return __builtin_amdgcn_wmma_f32_16x16x4_f32(false, a, false, b,
                                               (short)0, c, false, false);
}

__device__ __forceinline__ float sigmoidf_(float x) {
  return 1.0f / (1.0f + __expf(-x));
}

// ---------------------------------------------------------------------------
// Init: scalar features padded with zeros into state vectors.
// ---------------------------------------------------------------------------
__global__ __launch_bounds__(256) void init_kernel(
    const float* __restrict__ path_init, const float* __restrict__ link_init,
    const float* __restrict__ node_init, float* __restrict__ path_state,
    float* __restrict__ link_state, float* __restrict__ node_state) {
  int i = blockIdx.x * 256 + threadIdx.x;
  if (i < N_PATHS * PD) {
    int p = i >> 5, c = i & 31;
    path_state[i] = (c == 0) ? path_init[p]
                  : (c == 1) ? path_init[N_PATHS + p] : 0.0f;
  }
  if (i < N_LINKS * LD) {
    int l = i >> 5, c = i & 31;
    link_state[i] = (c == 0) ? link_init[l] : 0.0f;
  }
  if (i < N_NODES * ND) {
    int n = i >> 5, c = i & 31;
    node_state[i] = (c == 0) ? node_init[n] : 0.0f;
  }
}

// ---------------------------------------------------------------------------
// Pair-swizzle a [K,N] row-major matrix into [K/2, N] of float2:
//   out2[kk*N + col] = { in[(2kk)*N + col], in[(2kk+1)*N + col] }
// ---------------------------------------------------------------------------
__global__ __launch_bounds__(256) void swizzle_kernel(
    const float* __restrict__ in, float* __restrict__ out, int K, int N) {
  int i = blockIdx.x * 256 + threadIdx.x;
  int total = (K / 2) * N;
  if (i < total) {
    int kk = i / N, col = i - kk * N;
    v2f w;
    w.x = in[(2 * kk) * N + col];
    w.y = in[(2 * kk + 1) * N + col];
    ((v2f*)out)[i] = w;
  }
}

// ---------------------------------------------------------------------------
// GRU over PLEN=8 steps. One wave handles 16 paths; per step:
//   mx[16x96] = x[16x64] @ K   (A gathered from link/node rows, 16 k-tiles)
//   mh[16x96] = h[16x32] @ R   (A from wave-private LDS hidden, 8 k-tiles)
// Gates computed elementwise in the D-layout; hidden written back to LDS;
// step outputs scatter-added into agg[link] via global f32 atomics.
// ---------------------------------------------------------------------------
__global__ __launch_bounds__(256) void gru_kernel(
    const float* __restrict__ link_state, const float* __restrict__ node_state,
    float* __restrict__ path_state, float* __restrict__ agg,
    const float* __restrict__ gK, const float* __restrict__ gR,
    const float* __restrict__ gB, const int* __restrict__ l2p,
    const int* __restrict__ n2p, int do_agg) {
  __shared__ __align__(16) v2f sK2[32 * GN];   // 24 KB  (64 K-rows paired)
  __shared__ __align__(16) v2f sR2[16 * GN];   // 12 KB  (32 K-rows paired)
  __shared__ __align__(16) float sB[2 * GN];
  __shared__ __align__(16) float sH[8][16][PD]; // per-wave hidden, 16 KB

  for (int i = threadIdx.x; i < 32 * GN; i += 256) {
    int kk = i / GN, col = i - kk * GN;
    v2f w; w.x = gK[(2 * kk) * GN + col]; w.y = gK[(2 * kk + 1) * GN + col];
    sK2[i] = w;
  }
  for (int i = threadIdx.x; i < 16 * GN; i += 256) {
    int kk = i / GN, col = i - kk * GN;
    v2f w; w.x = gR[(2 * kk) * GN + col]; w.y = gR[(2 * kk + 1) * GN + col];
    sR2[i] = w;
  }
  for (int i = threadIdx.x; i < 2 * GN; i += 256) sB[i] = gB[i];

  const int wave = threadIdx.x >> 5;
  const int lane = threadIdx.x & 31;
  const int kh = lane >> 4;     // K-half select (A/B frag), row-half (D frag)
  const int mrow = lane & 15;   // A row / B,D column
  const int pbase = (blockIdx.x * 8 + wave) * 16;

  float(*h)[PD] = sH[wave];
  for (int i = lane; i < 16 * PD; i += 32)
    h[i >> 5][i & 31] = path_state[(size_t)(pbase + (i >> 5)) * PD + (i & 31)];
  __syncthreads();

  const v8f vz = {0.f, 0.f, 0.f, 0.f, 0.f, 0.f, 0.f, 0.f};

  for (int s = 0; s < PLEN; ++s) {
    const int e = (pbase + mrow) * PLEN + s;
    const float* xl = link_state + (size_t)l2p[e] * LD;
    const float* xn = node_state + (size_t)n2p[e] * ND;

    v8f mx[6], mh[6];
#pragma unroll
    for (int t = 0; t < 6; ++t) { mx[t] = vz; mh[t] = vz; }

    // x @ K : K-dim = 64 (link 0..31, node 32..63)
#pragma unroll 4
    for (int kt = 0; kt < 16; ++kt) {
      const int kb = kt * 4 + kh * 2;
      v2f a = (kb < LD) ? *(const v2f*)(xl + kb) : *(const v2f*)(xn + kb - LD);
      const v2f* brow = sK2 + (size_t)(kt * 2 + kh) * GN + mrow;
#pragma unroll
      for (int t = 0; t < 6; ++t) mx[t] = wmma4(a, brow[t * 16], mx[t]);
    }
    // h @ R : K-dim = 32
#pragma unroll 4
    for (int kt = 0; kt < 8; ++kt) {
      const int kb = kt * 4 + kh * 2;
      v2f a = *(const v2f*)&h[mrow][kb];
      const v2f* brow = sR2 + (size_t)(kt * 2 + kh) * GN + mrow;
#pragma unroll
      for (int t = 0; t < 6; ++t) mh[t] = wmma4(a, brow[t * 16], mh[t]);
    }
    // gates (Keras reset_after): D layout row = r + 8*kh, col = ct*16 + mrow
#pragma unroll
    for (int r = 0; r < 8; ++r) {
      const int row = r + 8 * kh;
      const int lrow = l2p[(pbase + row) * PLEN + s];
#pragma unroll
      for (int ct = 0; ct < 2; ++ct) {
        const int col = ct * 16 + mrow;
        float xz = mx[ct][r]     + sB[col];
        float xr = mx[2 + ct][r] + sB[PD + col];
        float xh = mx[4 + ct][r] + sB[2 * PD + col];
        float rz = mh[ct][r]     + sB[GN + col];
        float rr = mh[2 + ct][r] + sB[GN + PD + col];
        float rh = mh[4 + ct][r] + sB[GN + 2 * PD + col];
        float z  = sigmoidf_(xz + rz);
        float rg = sigmoidf_(xr + rr);
        float hh = tanhf(xh + rg * rh);
        float hn = z * h[row][col] + (1.0f - z) * hh;
        h[row][col] = hn;  // wave-private region; DS ops in-order per wave
        if (do_agg) atomicAdd(&agg[(size_t)lrow * LD + col], hn);
      }
    }
  }
  for (int i = lane; i < 16 * PD; i += 32)
    path_state[(size_t)(pbase + (i >> 5)) * PD + (i & 31)] = h[i >> 5][i & 31];
}

// ---------------------------------------------------------------------------
// Edge MLP: [link‖node(n2l)‖agg] (96) -> 32, then 4x (32 -> 32), ReLU each.
// One wave = 16 links; intermediates round-trip wave-private LDS.
// ---------------------------------------------------------------------------
__global__ __launch_bounds__(256) void edge_kernel(
    const float* __restrict__ link_in, const float* __restrict__ node_state,
    const float* __restrict__ aggv, const float* __restrict__ eW0,
    const float* __restrict__ eb0, const float* __restrict__ eWm,
    const float* __restrict__ ebm, const int* __restrict__ n2l,
    float* __restrict__ link_out) {
  __shared__ __align__(16) v2f sW02[48 * LD];    // 12 KB (96 K-rows paired)
  __shared__ __align__(16) v2f sWm2[4][16 * LD]; // 16 KB
  __shared__ __align__(16) float sE[8][16][LD];  // 16 KB

  for (int i = threadIdx.x; i < 48 * LD; i += 256) {
    int kk = i / LD, col = i - kk * LD;
    v2f w; w.x = eW0[(2 * kk) * LD + col]; w.y = eW0[(2 * kk + 1) * LD + col];
    sW02[i] = w;
  }
  for (int i = threadIdx.x; i < 4 * 16 * LD; i += 256) {
    int layer = i >> 9, j = i & 511;
    int kk = j / LD, col = j - kk * LD;
    const float* wm = eWm + layer * LD * LD;
    v2f w; w.x = wm[(2 * kk) * LD + col]; w.y = wm[(2 * kk + 1) * LD + col];
    sWm2[layer][j] = w;
  }

  const int wave = threadIdx.x >> 5;
  const int lane = threadIdx.x & 31;
  const int kh = lane >> 4;
  const int mrow = lane & 15;
  const int lbase = (blockIdx.x * 8 + wave) * 16;
  __syncthreads();

  const float* xl = link_in + (size_t)(lbase + mrow) * LD;
  const float* xn = node_state + (size_t)n2l[lbase + mrow] * ND;
  const float* xa = aggv + (size_t)(lbase + mrow) * LD;

  const v8f vz = {0.f, 0.f, 0.f, 0.f, 0.f, 0.f, 0.f, 0.f};
  v8f acc[2] = {vz, vz};
#pragma unroll 4
  for (int kt = 0; kt < 24; ++kt) {
    const int kb = kt * 4 + kh * 2;
    v2f a;
    if (kb < 32)      a = *(const v2f*)(xl + kb);
    else if (kb < 64) a = *(const v2f*)(xn + kb - 32);
    else              a = *(const v2f*)(xa + kb - 64);
    const v2f* brow = sW02 + (size_t)(kt * 2 + kh) * LD + mrow;
#pragma unroll
    for (int t = 0; t < 2; ++t) acc[t] = wmma4(a, brow[t * 16], acc[t]);
  }
  float(*eb)[LD] = sE[wave];
#pragma unroll
  for (int r = 0; r < 8; ++r)
#pragma unroll
    for (int ct = 0; ct < 2; ++ct) {
      const int col = ct * 16 + mrow;
      eb[r + 8 * kh][col] = fmaxf(acc[ct][r] + eb0[col], 0.0f);
    }

  for (int layer = 0; layer < 4; ++layer) {
    acc[0] = vz; acc[1] = vz;
#pragma unroll
    for (int kt = 0; kt < 8; ++kt) {
      const int kb = kt * 4 + kh * 2;
      v2f a = *(const v2f*)&eb[mrow][kb];
      const v2f* brow = sWm2[layer] + (size_t)(kt * 2 + kh) * LD + mrow;
#pragma unroll
      for (int t = 0; t < 2; ++t) acc[t] = wmma4(a, brow[t * 16], acc[t]);
    }
#pragma unroll
    for (int r = 0; r < 8; ++r)
#pragma unroll
      for (int ct = 0; ct < 2; ++ct) {
        const int col = ct * 16 + mrow;
        eb[r + 8 * kh][col] = fmaxf(acc[ct][r] + ebm[layer * LD + col], 0.0f);
      }
  }
  for (int i = lane; i < 16 * LD; i += 32)
    link_out[(size_t)(lbase + (i >> 5)) * LD + (i & 31)] = eb[i >> 5][i & 31];
}

// ---------------------------------------------------------------------------
// Node update stage A: tmp = node_state @ nW   [4096,32]x[32,32]
// tmp is written PAIR-SWIZZLED: tmp2[(row/2)*ND + col] = {t[row0], t[row1]}
// so node_adj can consume it as B fragments with single b64 loads.
// ---------------------------------------------------------------------------
__global__ __launch_bounds__(256) void node_mm_kernel(
    const float* __restrict__ node_state, const float* __restrict__ nW,
    float* __restrict__ tmp) {
  __shared__ __align__(16) v2f sNW2[16 * ND];  // 4 KB
  for (int i = threadIdx.x; i < 16 * ND; i += 256) {
    int kk = i / ND, col = i - kk * ND;
    v2f w; w.x = nW[(2 * kk) * ND + col]; w.y = nW[(2 * kk + 1) * ND + col];
    sNW2[i] = w;
  }
  const int wave = threadIdx.x >> 5, lane = threadIdx.x & 31;
  const int kh = lane >> 4, mrow = lane & 15;
  const int nbase = (blockIdx.x * 8 + wave) * 16;
  __syncthreads();
  const float* x = node_state + (size_t)(nbase + mrow) * ND;
  const v8f vz = {0.f, 0.f, 0.f, 0.f, 0.f, 0.f, 0.f, 0.f};
  v8f acc[2] = {vz, vz};
#pragma unroll
  for (int kt = 0; kt < 8; ++kt) {
    const int kb = kt * 4 + kh * 2;
    v2f a = *(const v2f*)(x + kb);
    const v2f* brow = sNW2 + (size_t)(kt * 2 + kh) * ND + mrow;
#pragma unroll
    for (int t = 0; t < 2; ++t) acc[t] = wmma4(a, brow[t * 16], acc[t]);
  }
#pragma unroll
  for (int r = 0; r < 8; ++r)
#pragma unroll
    for (int ct = 0; ct < 2; ++ct) {
      const int row = nbase + r + 8 * kh;
      const int col = ct * 16 + mrow;
      // swizzled address: element [row][col] -> ((row>>1)*ND + col)*2 + (row&1)
      tmp[(((size_t)(row >> 1) * ND + col) << 1) + (row & 1)] = acc[ct][r];
    }
}

// ---------------------------------------------------------------------------
// Node update stage B: node_out = relu(adj @ tmp + nb)  [4096,4096]x[4096,32]
// ---------------------------------------------------------------------------
__global__ __launch_bounds__(256) void node_adj_kernel(
    const float* __restrict__ adj, const float* __restrict__ tmp,
    const float* __restrict__ nb, float* __restrict__ node_out) {
  const int wave = threadIdx.x >> 5, lane = threadIdx.x & 31;
  const int kh = lane >> 4, mrow = lane & 15;
  const int nbase = (blockIdx.x * 8 + wave) * 16;
  const float* arow = adj + (size_t)(nbase + mrow) * N_NODES;
  const v2f* tmpv = (const v2f*)tmp;  // pair-swizzled [K/2][ND]
  const v8f vz = {0.f, 0.f, 0.f, 0.f, 0.f, 0.f, 0.f, 0.f};
  v8f acc[2] = {vz, vz};
#pragma unroll 4
  for (int kt = 0; kt < N_NODES / 4; ++kt) {
    const int kb = kt * 4 + kh * 2;
    v2f a = *(const v2f*)(arow + kb);
    const v2f* brow = tmpv + (size_t)(kt * 2 + kh) * ND + mrow;
#pragma unroll
    for (int t = 0; t < 2; ++t) acc[t] = wmma4(a, brow[t * 16], acc[t]);
  }
#pragma unroll
  for (int r = 0; r < 8; ++r)
#pragma unroll
    for (int ct = 0; ct < 2; ++ct) {
      const int col = ct * 16 + mrow;
      node_out[(size_t)(nbase + r + 8 * kh) * ND + col] =
          fmaxf(acc[ct][r] + nb[col], 0.0f);
    }
}

// ---------------------------------------------------------------------------
// Fused readout: r0 = relu(ps@rW0+rb0); r1 = relu(r0@rW1+rb1);
// out = ps@fW[:32] + r1@fW[32:] + fb. r1 never materialized; per-lane
// partials reduced across the 16-lane half via shuffles.
// rW0/rW1 consumed pair-swizzled from workspace.
// ---------------------------------------------------------------------------
__global__ __launch_bounds__(128) void readout_kernel(
    const float* __restrict__ path_state, const float* __restrict__ rW0s,
    const float* __restrict__ rb0, const float* __restrict__ rW1s,
    const float* __restrict__ rb1, const float* __restrict__ fW,
    const float* __restrict__ fb, float* __restrict__ out) {
  __shared__ __align__(16) float sR0[4][16][RU];  // 64 KB, wave-private
  const int wave = threadIdx.x >> 5, lane = threadIdx.x & 31;
  const int kh = lane >> 4, mrow = lane & 15;
  const int pbase = (blockIdx.x * 4 + wave) * 16;
  const float* ps = path_state + (size_t)pbase * PD;
  const v2f* w0v = (const v2f*)rW0s;  // [(PD/2)][RU]
  const v2f* w1v = (const v2f*)rW1s;  // [(RU/2)][RU]
  float(*r0)[RU] = sR0[wave];
  const v8f vz = {0.f, 0.f, 0.f, 0.f, 0.f, 0.f, 0.f, 0.f};

  // layer 0: 16x32 @ 32x256
  for (int nt = 0; nt < 16; ++nt) {
    const int col = nt * 16 + mrow;
    v8f acc = vz;
#pragma unroll
    for (int kt = 0; kt < 8; ++kt) {
      const int kb = kt * 4 + kh * 2;
      v2f a = *(const v2f*)(ps + mrow * PD + kb);
      acc = wmma4(a, w0v[(size_t)(kt * 2 + kh) * RU + col], acc);
    }
#pragma unroll
    for (int r = 0; r < 8; ++r)
      r0[r + 8 * kh][col] = fmaxf(acc[r] + rb0[col], 0.0f);
  }

  // layer 1 (16x256 @ 256x256) fused with final dot against fW[32:]
  float op[8];
#pragma unroll
  for (int r = 0; r < 8; ++r) op[r] = 0.0f;
  for (int nt = 0; nt < 16; ++nt) {
    const int col = nt * 16 + mrow;
    v8f acc = vz;
#pragma unroll 8
    for (int kt = 0; kt < 64; ++kt) {
      const int kb = kt * 4 + kh * 2;
      v2f a = *(const v2f*)&r0[mrow][kb];
      acc = wmma4(a, w1v[(size_t)(kt * 2 + kh) * RU + col], acc);
    }
    const float w = fW[PD + col];
#pragma unroll
    for (int r = 0; r < 8; ++r)
      op[r] += fmaxf(acc[r] + rb1[col], 0.0f) * w;
  }
  // path_state direct contribution (cols split across the 16 lanes)
#pragma unroll
  for (int r = 0; r < 8; ++r) {
    const int row = r + 8 * kh;
    op[r] += ps[row * PD + mrow] * fW[mrow] +
             ps[row * PD + 16 + mrow] * fW[16 + mrow];
  }
  // reduce across the 16 lanes of each half-wave
#pragma unroll
  for (int off = 8; off >= 1; off >>= 1)
#pragma unroll
    for (int r = 0; r < 8; ++r) op[r] += __shfl_xor(op[r], off, 16);
  if (mrow == 0) {
#pragma unroll
    for (int r = 0; r < 8; ++r) out[pbase + r + 8 * kh] = op[r] + fb[0];
  }
}

// ---------------------------------------------------------------------------
extern "C" void kernel_launch(void* const* d_in, const int* in_sizes, int n_in,
                              void* d_out, int out_size, void* d_ws,
                              size_t ws_size, hipStream_t stream) {
  const float* path_init = (const float*)d_in[0];
  const float* link_init = (const float*)d_in[1];
  const float* node_init = (const float*)d_in[2];
  const float* adj    = (const float*)d_in[3];
  const float* gru_k  = (const float*)d_in[4];
  const float* gru_rk = (const float*)d_in[5];
  const float* gru_b  = (const float*)d_in[6];
  const float* eW0    = (const float*)d_in[7];
  const float* eb0    = (const float*)d_in[8];
  const float* eWm    = (const float*)d_in[9];
  const float* ebm    = (const float*)d_in[10];
  const float* nW     = (const float*)d_in[11];
  const float* nb     = (const float*)d_in[12];
  const float* rW0    = (const float*)d_in[13];
  const float* rb0    = (const float*)d_in[14];
  const float* rW1    = (const float*)d_in[15];
  const float* rb1    = (const float*)d_in[16];
  const float* fW     = (const float*)d_in[17];
  const float* fb     = (const float*)d_in[18];
  // d_in[19]=paths, d_in[20]=seqs are identity reshapes -> unused
  const int* l2p = (const int*)d_in[21];
  const int* n2p = (const int*)d_in[22];
  const int* n2l = (const int*)d_in[23];

  char* ws = (char*)d_ws;
  float* path_state = (float*)ws; ws += (size_t)N_PATHS * PD * 4;
  float* link_a = (float*)ws;     ws += (size_t)N_LINKS * LD * 4;
  float* link_b = (float*)ws;     ws += (size_t)N_LINKS * LD * 4;
  float* node_a = (float*)ws;     ws += (size_t)N_NODES * ND * 4;
  float* node_b = (float*)ws;     ws += (size_t)N_NODES * ND * 4;
  float* agg    = (float*)ws;     ws += (size_t)N_LINKS * LD * 4;
  float* tmpn   = (float*)ws;     ws += (size_t)N_NODES * ND * 4;
  float* rW0s   = (float*)ws;     ws += (size_t)PD * RU * 4;
  float* rW1s   = (float*)ws;     ws += (size_t)RU * RU * 4;

  init_kernel<<<(N_PATHS * PD + 255) / 256, 256, 0, stream>>>(
      path_init, link_init, node_init, path_state, link_a, node_a);
  swizzle_kernel<<<((PD / 2) * RU + 255) / 256, 256, 0, stream>>>(rW0, rW0s,
                                                                  PD, RU);
  swizzle_kernel<<<((RU / 2) * RU + 255) / 256, 256, 0, stream>>>(rW1, rW1s,
                                                                  RU, RU);

  float* link_cur = link_a; float* link_nxt = link_b;
  float* node_cur = node_a; float* node_nxt = node_b;

  for (int ti = 0; ti < 4; ++ti) {
    const int do_agg = (ti < 3);
    if (do_agg)
      hipMemsetAsync(agg, 0, (size_t)N_LINKS * LD * 4, stream);
    gru_kernel<<<N_PATHS / (16 * 8), 256, 0, stream>>>(
        link_cur, node_cur, path_state, agg, gru_k, gru_rk, gru_b, l2p, n2p,
        do_agg);
    if (do_agg) {
      edge_kernel<<<N_LINKS / (16 * 8), 256, 0, stream>>>(
          link_cur, node_cur, agg, eW0, eb0, eWm, ebm, n2l, link_nxt);
      node_mm_kernel<<<N_NODES / (16 * 8), 256, 0, stream>>>(node_cur, nW,
                                                             tmpn);
      node_adj_kernel<<<N_NODES / (16 * 8), 256, 0, stream>>>(adj, tmpn, nb,
                                                              node_nxt);
      float* t;
      t = link_cur; link_cur = link_nxt; link_nxt = t;
      t = node_cur; node_cur = node_nxt; node_nxt = t;
    }
  }

  readout_kernel<<<N_PATHS / (16 * 4), 128, 0, stream>>>(
      path_state, rW0s, rb0, rW1s, rb1, fW, fb, (float*)d_out);
}